// GraphConvolution_nBatch_78950088835519
// MI455X (gfx1250) — compile-verified
//
#include <hip/hip_runtime.h>
#include <hip/hip_bf16.h>

typedef __attribute__((ext_vector_type(2))) float v2f;
typedef __attribute__((ext_vector_type(8))) float v8f;

#define IN_F  256
#define OUT_F 256
#define LDS_STRIDE 260   // 260 % 64 == 4 -> conflict-free b64 A-fragment reads

// ---------------------------------------------------------------------------
// Kernel 0: transpose weight [IN_F][OUT_F] -> Wt [OUT_F][IN_F] (one-time 256KB)
// ---------------------------------------------------------------------------
__global__ void transpose256(const float* __restrict__ W, float* __restrict__ Wt) {
    __shared__ float tile[32][33];
    const int bx = blockIdx.x * 32;   // column block of W
    const int by = blockIdx.y * 32;   // row block of W
    const int tx = threadIdx.x;       // 0..31
    const int ty = threadIdx.y;       // 0..7
#pragma unroll
    for (int i = 0; i < 32; i += 8)
        tile[ty + i][tx] = W[(size_t)(by + ty + i) * OUT_F + bx + tx];
    __syncthreads();
#pragma unroll
    for (int i = 0; i < 32; i += 8)
        Wt[(size_t)(bx + ty + i) * IN_F + by + tx] = tile[tx][ty + i];
}

// ---------------------------------------------------------------------------
// Kernel 1: support = x @ W via V_WMMA_F32_16X16X4_F32.
// Block: 256 threads = 8 waves; handles 64 rows x 256 cols.
// Wave w owns col-tiles {2w, 2w+1} across 4 row-tiles -> 8 C accumulators.
// ---------------------------------------------------------------------------
__global__ __launch_bounds__(256, 2)
void gcn_gemm_wmma(const float* __restrict__ x, const float* __restrict__ Wt,
                   float* __restrict__ support, int N) {
    __shared__ float xs[64 * LDS_STRIDE];

    const int tid      = threadIdx.x;
    const int blockRow = blockIdx.x * 64;

    // Cooperative load of the 64x256 x-strip into LDS (float4 global reads,
    // float2 LDS stores to keep 8B alignment with stride 260).
    for (int idx = tid; idx < 64 * 64; idx += 256) {
        const int r  = idx >> 6;        // strip row 0..63
        const int c4 = idx & 63;        // float4 column 0..63
        const int g  = blockRow + r;
        float4 v = make_float4(0.f, 0.f, 0.f, 0.f);
        if (g < N) v = ((const float4*)(x + (size_t)g * IN_F))[c4];
        float* p = &xs[r * LDS_STRIDE + c4 * 4];
        ((float2*)p)[0] = make_float2(v.x, v.y);
        ((float2*)p)[1] = make_float2(v.z, v.w);
    }
    __syncthreads();

    const int wave = tid >> 5;
    const int lane = tid & 31;
    const int half = lane >> 4;   // 0: lanes 0-15, 1: lanes 16-31
    const int lrow = lane & 15;

    v8f acc[4][2];
#pragma unroll
    for (int rt = 0; rt < 4; ++rt)
#pragma unroll
        for (int j = 0; j < 2; ++j)
#pragma unroll
            for (int i = 0; i < 8; ++i) acc[rt][j][i] = 0.0f;

    // B fragment source rows in transposed weight: Wt[col][k] contiguous in k.
    const float* wt0 = Wt + (size_t)((wave * 2 + 0) * 16 + lrow) * IN_F;
    const float* wt1 = Wt + (size_t)((wave * 2 + 1) * 16 + lrow) * IN_F;

    for (int k0 = 0; k0 < IN_F; k0 += 4) {
        const int ks = k0 + 2 * half;    // A/B layout: lanes 16-31 hold K+2,K+3
        const v2f b0 = *(const v2f*)(wt0 + ks);
        const v2f b1 = *(const v2f*)(wt1 + ks);
#pragma unroll
        for (int rt = 0; rt < 4; ++rt) {
            const v2f a = *(const v2f*)&xs[(rt * 16 + lrow) * LDS_STRIDE + ks];
            acc[rt][0] = __builtin_amdgcn_wmma_f32_16x16x4_f32(
                false, a, false, b0, (short)0, acc[rt][0], false, false);
            acc[rt][1] = __builtin_amdgcn_wmma_f32_16x16x4_f32(
                false, a, false, b1, (short)0, acc[rt][1], false, false);
        }
    }

    // C/D layout: VGPR i -> M = i + 8*half, N = lrow (within a 16x16 tile).
#pragma unroll
    for (int rt = 0; rt < 4; ++rt) {
#pragma unroll
        for (int j = 0; j < 2; ++j) {
            const int col = (wave * 2 + j) * 16 + lrow;
#pragma unroll
            for (int i = 0; i < 8; ++i) {
                const int g = blockRow + rt * 16 + i + 8 * half;
                if (g < N) support[(size_t)g * OUT_F + col] = acc[rt][j][i];
            }
        }
    }
}

// ---------------------------------------------------------------------------
// Kernel 2: out[n][f] = bias[f]  (harness poisons d_out; float4 stores)
// ---------------------------------------------------------------------------
__global__ void init_bias(float4* __restrict__ out4, const float4* __restrict__ bias4,
                          long long total4) {
    const long long i = (long long)blockIdx.x * blockDim.x + threadIdx.x;
    if (i < total4) out4[i] = bias4[i & 63];   // OUT_F/4 == 64
}

// ---------------------------------------------------------------------------
// Kernel 3: edge-parallel scatter SpMM.
// Each wave32 loads 32 edges' metadata coalesced, shfl-broadcasts one edge at
// a time; 32 lanes x 8 floats cover the 256-feature row.
// support and out are L2-resident (102MB each < 192MB L2).
// ---------------------------------------------------------------------------
__global__ __launch_bounds__(256)
void spmm_scatter(const long long* __restrict__ ei, const float* __restrict__ ew,
                  const float* __restrict__ support, float* __restrict__ out,
                  int E, int N) {
    const int wave = (int)((blockIdx.x * blockDim.x + threadIdx.x) >> 5);
    const int lane = threadIdx.x & 31;
    const long long e0 = (long long)wave * 32;
    if (e0 >= E) return;

    int   myRow = 0, myCol = 0;
    float myW   = 0.0f;
    if (e0 + lane < E) {
        myRow = (int)ei[e0 + lane];
        myCol = (int)ei[(long long)E + e0 + lane];
        myW   = ew[e0 + lane];
    }

    for (int i = 0; i < 32; ++i) {
        const int   r  = __shfl(myRow, i, 32);
        const int   c  = __shfl(myCol, i, 32);
        const float wt = __shfl(myW,  i, 32);

        const float4* src = (const float4*)(support + (size_t)c * OUT_F) + lane * 2;
        const float4 s0 = src[0];
        const float4 s1 = src[1];

        float* dst = out + (size_t)r * OUT_F + lane * 8;
        __hip_atomic_fetch_add(dst + 0, wt * s0.x, __ATOMIC_RELAXED, __HIP_MEMORY_SCOPE_AGENT);
        __hip_atomic_fetch_add(dst + 1, wt * s0.y, __ATOMIC_RELAXED, __HIP_MEMORY_SCOPE_AGENT);
        __hip_atomic_fetch_add(dst + 2, wt * s0.z, __ATOMIC_RELAXED, __HIP_MEMORY_SCOPE_AGENT);
        __hip_atomic_fetch_add(dst + 3, wt * s0.w, __ATOMIC_RELAXED, __HIP_MEMORY_SCOPE_AGENT);
        __hip_atomic_fetch_add(dst + 4, wt * s1.x, __ATOMIC_RELAXED, __HIP_MEMORY_SCOPE_AGENT);
        __hip_atomic_fetch_add(dst + 5, wt * s1.y, __ATOMIC_RELAXED, __HIP_MEMORY_SCOPE_AGENT);
        __hip_atomic_fetch_add(dst + 6, wt * s1.z, __ATOMIC_RELAXED, __HIP_MEMORY_SCOPE_AGENT);
        __hip_atomic_fetch_add(dst + 7, wt * s1.w, __ATOMIC_RELAXED, __HIP_MEMORY_SCOPE_AGENT);
    }
}

// ---------------------------------------------------------------------------
extern "C" void kernel_launch(void* const* d_in, const int* in_sizes, int n_in,
                              void* d_out, int out_size, void* d_ws, size_t ws_size,
                              hipStream_t stream) {
    const float*     x    = (const float*)d_in[0];
    const long long* ei   = (const long long*)d_in[1];
    const float*     ew   = (const float*)d_in[2];
    const float*     W    = (const float*)d_in[3];
    const float*     bias = (const float*)d_in[4];
    float*           out  = (float*)d_out;

    const int N = in_sizes[0] / IN_F;   // 100000
    const int E = in_sizes[2];          // 3200000

    // Workspace layout: [support: N*OUT_F floats][Wt: IN_F*OUT_F floats]
    float* support = (float*)d_ws;
    float* Wt      = support + (size_t)N * OUT_F;

    // 0) transpose weight (256x256)
    transpose256<<<dim3(8, 8), dim3(32, 8), 0, stream>>>(W, Wt);

    // 1) support = x @ W  (WMMA fp32)
    const int gemmBlocks = (N + 63) / 64;
    gcn_gemm_wmma<<<gemmBlocks, 256, 0, stream>>>(x, Wt, support, N);

    // 2) out = bias broadcast
    const long long total4 = (long long)N * (OUT_F / 4);
    init_bias<<<(int)((total4 + 255) / 256), 256, 0, stream>>>(
        (float4*)out, (const float4*)bias, total4);

    // 3) scatter-add over edges
    const int waves   = (E + 31) / 32;
    const int sBlocks = (waves + 7) / 8;   // 8 waves per 256-thread block
    spmm_scatter<<<sBlocks, 256, 0, stream>>>(ei, ew, support, out, E, N);
}